// DeadPixelCorrection_54546084659556
// MI455X (gfx1250) — compile-verified
//
#include <hip/hip_runtime.h>
#include <stdint.h>

// Problem constants (fixed by the reference: B=4, C=1, H=2160, W=3840, k=3, thr=30)
#define B_   4
#define H_   2160
#define W_   3840
#define TH   8          // output rows per block
#define TW   128        // output cols per block
#define NTHREADS 256    // 8 wave32 waves
#define LDS_STRIDE 136  // floats per LDS row: halo-left at 3, interior 4..131, halo-right 132
#define THRESH 30.0f

typedef float v4f __attribute__((ext_vector_type(4)));  // native clang vector (16B)

// gfx1250 async global->LDS path (ASYNCcnt-tracked), with safe fallback.
#if defined(__gfx1250__) && \
    __has_builtin(__builtin_amdgcn_global_load_async_to_lds_b128) && \
    __has_builtin(__builtin_amdgcn_s_wait_asynccnt)
#define USE_ASYNC_LDS 1
typedef int v4i_ __attribute__((vector_size(16)));
typedef __attribute__((address_space(1))) v4i_ gv4i;   // global int4
typedef __attribute__((address_space(3))) v4i_ lv4i;   // LDS int4
#else
#define USE_ASYNC_LDS 0
#endif

__device__ __forceinline__ float med3f(float x, float y, float z) {
    // median of 3 = max(min(x,y), min(max(x,y), z))
    return fmaxf(fminf(x, y), fminf(fmaxf(x, y), z));
}

__device__ __forceinline__ int reflect_row(int y) {
    y = (y < 0) ? -y : y;
    y = (y >= H_) ? (2 * H_ - 2 - y) : y;
    return y;
}

__global__ __launch_bounds__(NTHREADS)
void dpc_median3x3_kernel(const float* __restrict__ img, float* __restrict__ out) {
    __shared__ float tile[(TH + 2) * LDS_STRIDE];

    const int tid = threadIdx.x;
    const int x0  = blockIdx.x * TW;
    const int y0  = blockIdx.y * TH;
    const int b   = blockIdx.z;
    const float* src = img + (size_t)b * H_ * W_;

    // ---------------- stage (TH+2) x (TW+2) reflect-padded tile into LDS ----------------
    // Interior: 10 rows x 32 float4 groups = 320 b128 transfers (perfectly coalesced),
    // issued on the gfx1250 ASYNCcnt-tracked global->LDS path.
    for (int t = tid; t < (TH + 2) * (TW / 4); t += NTHREADS) {
        const int lr = t >> 5;           // lds row 0..9
        const int g4 = (t & 31) << 2;    // float offset within row, multiple of 4
        const int gy = reflect_row(y0 - 1 + lr);
        const float* gp = src + (size_t)gy * W_ + (x0 + g4);
        float* lp = &tile[lr * LDS_STRIDE + 4 + g4];  // byte addr ≡ 0 mod 16
#if USE_ASYNC_LDS
        __builtin_amdgcn_global_load_async_to_lds_b128(
            (gv4i*)gp, (lv4i*)lp, 0, 0);
#else
        *(v4f*)lp = *(const v4f*)gp;
#endif
    }
    // Halo columns: 10 rows x {left,right} = 20 scalars (reflected at image borders).
    if (tid < (TH + 2) * 2) {
        const int lr = tid >> 1, side = tid & 1;
        const int gy = reflect_row(y0 - 1 + lr);
        int gx = side ? (x0 + TW) : (x0 - 1);
        gx = (gx < 0) ? 1 : ((gx >= W_) ? (W_ - 2) : gx);
        tile[lr * LDS_STRIDE + (side ? (4 + TW) : 3)] = src[(size_t)gy * W_ + gx];
    }
#if USE_ASYNC_LDS
    __builtin_amdgcn_s_wait_asynccnt(0);
#endif
    __syncthreads();

    // ---------------- compute: 4 output pixels per thread ----------------
    const int r = tid >> 5;          // output row in tile: 0..7
    const int c = (tid & 31) << 2;   // output col group:  0,4,...,124

    // Need lds cols c+3 .. c+8 (centers at c+4..c+7) on lds rows r, r+1, r+2.
    float a0[6], a1[6], a2[6];
    {
        const float* p0 = &tile[(r + 0) * LDS_STRIDE + c];
        const float* p1 = &tile[(r + 1) * LDS_STRIDE + c];
        const float* p2 = &tile[(r + 2) * LDS_STRIDE + c];
        v4f m;
        m = *(const v4f*)(p0 + 4);
        a0[0] = p0[3]; a0[1] = m.x; a0[2] = m.y; a0[3] = m.z; a0[4] = m.w; a0[5] = p0[8];
        m = *(const v4f*)(p1 + 4);
        a1[0] = p1[3]; a1[1] = m.x; a1[2] = m.y; a1[3] = m.z; a1[4] = m.w; a1[5] = p1[8];
        m = *(const v4f*)(p2 + 4);
        a2[0] = p2[3]; a2[1] = m.x; a2[2] = m.y; a2[3] = m.z; a2[4] = m.w; a2[5] = p2[8];
    }

    // Vertical sort3 of each of the 6 columns (shared across the 4 outputs).
    float lo[6], mi[6], hi[6];
#pragma unroll
    for (int k = 0; k < 6; ++k) {
        const float mn01 = fminf(a0[k], a1[k]);
        const float mx01 = fmaxf(a0[k], a1[k]);
        lo[k] = fminf(mn01, a2[k]);
        hi[k] = fmaxf(mx01, a2[k]);
        mi[k] = fmaxf(mn01, fminf(mx01, a2[k]));
    }

    v4f res;
#pragma unroll
    for (int j = 0; j < 4; ++j) {
        const float mx  = fmaxf(fmaxf(lo[j], lo[j + 1]), lo[j + 2]);
        const float mn  = fminf(fminf(hi[j], hi[j + 1]), hi[j + 2]);
        const float md  = med3f(mi[j], mi[j + 1], mi[j + 2]);
        const float med = med3f(mx, md, mn);        // exact median of the 9
        const float cen = a1[j + 1];                // original center pixel
        const float d   = fabsf(cen - med);
        res[j] = (d > THRESH) ? med : cen;          // values are exact small ints
    }

    const size_t o = ((size_t)b * H_ + (size_t)(y0 + r)) * (size_t)W_ + (size_t)(x0 + c);
    // Output is write-once: non-temporal store keeps L2 capacity for the input
    // stream (whose rows are re-read as halo by vertically adjacent blocks).
    __builtin_nontemporal_store(res, (v4f*)(out + o));
}

extern "C" void kernel_launch(void* const* d_in, const int* in_sizes, int n_in,
                              void* d_out, int out_size, void* d_ws, size_t ws_size,
                              hipStream_t stream) {
    (void)in_sizes; (void)n_in; (void)d_ws; (void)ws_size; (void)out_size;
    const float* bayer = (const float*)d_in[0];
    float* out = (float*)d_out;
    dim3 grid(W_ / TW, H_ / TH, B_);
    dpc_median3x3_kernel<<<grid, NTHREADS, 0, stream>>>(bayer, out);
}